// CustomSAGPooling_6416681140433
// MI455X (gfx1250) — compile-verified
//
#include <hip/hip_runtime.h>
#include <hip/hip_bf16.h>
#include <math.h>

typedef __attribute__((ext_vector_type(2))) float v2f;
typedef __attribute__((ext_vector_type(8))) float v8f;

#define TB 256
#define D 128

// ---------------- Stage 1: per-node dots p = w_rel_in.x, r = w_root_in.x ----
__global__ void k_node_dots(const float* __restrict__ x,
                            const float* __restrict__ w_rel_in,
                            const float* __restrict__ w_root_in,
                            float* __restrict__ p, float* __restrict__ r,
                            float* __restrict__ attnsum, float* __restrict__ deg,
                            int N) {
  int wave = (int)(((size_t)blockIdx.x * blockDim.x + threadIdx.x) >> 5);
  int lane = threadIdx.x & 31;
  if (wave >= N) return;                       // wave-uniform exit
  float4 xa = ((const float4*)(x + (size_t)wave * D))[lane];
  float4 wa = ((const float4*)w_rel_in)[lane];
  float4 wb = ((const float4*)w_root_in)[lane];
  float pr = xa.x*wa.x + xa.y*wa.y + xa.z*wa.z + xa.w*wa.w;
  float rr = xa.x*wb.x + xa.y*wb.y + xa.z*wb.z + xa.w*wb.w;
#pragma unroll
  for (int off = 16; off > 0; off >>= 1) {
    pr += __shfl_xor(pr, off, 32);
    rr += __shfl_xor(rr, off, 32);
  }
  if (lane == 0) { p[wave] = pr; r[wave] = rr; attnsum[wave] = 0.f; deg[wave] = 0.f; }
}

// ---------------- Stage 2: scalar edge scatter (factorized GraphConv D->1) --
__global__ void k_edge_scatter_in(const long long* __restrict__ ei,
                                  const float* __restrict__ p,
                                  float* __restrict__ attnsum,
                                  float* __restrict__ deg, int E) {
  int e = blockIdx.x * blockDim.x + threadIdx.x;
  if (e >= E) return;
  int s = (int)ei[e];
  int d = (int)ei[e + E];
  atomicAdd(&attnsum[d], p[s]);
  atomicAdd(&deg[d], 1.0f);
}

// ---------------- Stage 3: attn = agg + b + root ; block max -----------------
__global__ void k_attn_max(const float* __restrict__ attnsum,
                           const float* __restrict__ r,
                           const float* __restrict__ b_in,
                           float* __restrict__ attn,
                           float* __restrict__ partmax, int N) {
  __shared__ float sm[TB];
  int i = blockIdx.x * blockDim.x + threadIdx.x;
  float v = -3.402823466e38f;
  if (i < N) { v = attnsum[i] + r[i] + b_in[0]; attn[i] = v; }
  sm[threadIdx.x] = v;
  __syncthreads();
  for (int s2 = TB / 2; s2 > 0; s2 >>= 1) {
    if ((int)threadIdx.x < s2) sm[threadIdx.x] = fmaxf(sm[threadIdx.x], sm[threadIdx.x + s2]);
    __syncthreads();
  }
  if (threadIdx.x == 0) partmax[blockIdx.x] = sm[0];
}

__global__ void k_reduce_final(const float* __restrict__ part, float* __restrict__ out,
                               int L, int isMax) {
  __shared__ float sm[TB];
  float acc = isMax ? -3.402823466e38f : 0.f;
  for (int i = threadIdx.x; i < L; i += TB)
    acc = isMax ? fmaxf(acc, part[i]) : (acc + part[i]);
  sm[threadIdx.x] = acc;
  __syncthreads();
  for (int s2 = TB / 2; s2 > 0; s2 >>= 1) {
    if ((int)threadIdx.x < s2)
      sm[threadIdx.x] = isMax ? fmaxf(sm[threadIdx.x], sm[threadIdx.x + s2])
                              : (sm[threadIdx.x] + sm[threadIdx.x + s2]);
    __syncthreads();
  }
  if (threadIdx.x == 0) out[0] = sm[0];
}

// ---------------- Stage 4: exp(attn-max), block sum --------------------------
__global__ void k_exp_sum(float* __restrict__ attn, const float* __restrict__ gmax,
                          float* __restrict__ partsum, int N) {
  __shared__ float sm[TB];
  int i = blockIdx.x * blockDim.x + threadIdx.x;
  float v = 0.f;
  if (i < N) { v = __expf(attn[i] - gmax[0]); attn[i] = v; }
  sm[threadIdx.x] = v;
  __syncthreads();
  for (int s2 = TB / 2; s2 > 0; s2 >>= 1) {
    if ((int)threadIdx.x < s2) sm[threadIdx.x] += sm[threadIdx.x + s2];
    __syncthreads();
  }
  if (threadIdx.x == 0) partsum[blockIdx.x] = sm[0];
}

// ---------------- Stage 5: score + sortable keys + node_map init -------------
__global__ void k_score_keys(const float* __restrict__ attn, const float* __restrict__ deg,
                             const float* __restrict__ gsum, const float* __restrict__ w_sel,
                             float* __restrict__ s, unsigned long long* __restrict__ keys,
                             int* __restrict__ node_map, int N, int SORTN) {
  int i = blockIdx.x * blockDim.x + threadIdx.x;
  if (i >= SORTN) return;
  if (i < N) {
    float soft = attn[i] / gsum[0];
    float sgn  = (w_sel[0] < 0.f) ? -1.f : 1.f;
    float sc   = tanhf(soft * deg[i] * sgn);
    s[i] = sc;
    node_map[i] = -1;
    unsigned u   = __float_as_uint(sc);
    unsigned ord = (u & 0x80000000u) ? ~u : (u | 0x80000000u); // ascending-order map
    unsigned flip = ~ord;                                      // ascending sort => descending score
    keys[i] = ((unsigned long long)flip << 32) | (unsigned)i;  // ties: smaller index first
  } else {
    keys[i] = 0xFFFFFFFFFFFFFFFFull;                           // pad sorts to end
  }
}

// ---------------- Stage 6: global bitonic sort pass --------------------------
__global__ void k_bitonic(unsigned long long* __restrict__ keys, int j, int sz) {
  int i = blockIdx.x * blockDim.x + threadIdx.x;
  int ixj = i ^ j;
  if (ixj > i) {
    unsigned long long a = keys[i], b = keys[ixj];
    bool asc = ((i & sz) == 0);
    if ((a > b) == asc) { keys[i] = b; keys[ixj] = a; }
  }
}

// ---------------- Stage 7: emit perm/score/batch, build node_map -------------
__global__ void k_select(const unsigned long long* __restrict__ keys,
                         const float* __restrict__ s,
                         int* __restrict__ node_map, int* __restrict__ perm_i,
                         float* __restrict__ out_perm, float* __restrict__ out_score,
                         float* __restrict__ out_batch, int k) {
  int t = blockIdx.x * blockDim.x + threadIdx.x;
  if (t >= k) return;
  int idx = (int)(keys[t] & 0xFFFFFFFFu);
  node_map[idx] = t;
  perm_i[t] = idx;
  out_perm[t]  = (float)idx;
  out_score[t] = s[idx];
  out_batch[t] = 0.f;
}

// ---------------- Stage 8: x_p = x[perm]*score ; zero agg2 -------------------
__global__ void k_xp(const float* __restrict__ x, const int* __restrict__ perm_i,
                     const float* __restrict__ s, float* __restrict__ xp,
                     float* __restrict__ agg2, int k) {
  int wave = (int)(((size_t)blockIdx.x * blockDim.x + threadIdx.x) >> 5);
  int lane = threadIdx.x & 31;
  if (wave >= k) return;
  int src = perm_i[wave];
  float sc = s[src];
  float4 v = ((const float4*)(x + (size_t)src * D))[lane];
  v.x *= sc; v.y *= sc; v.z *= sc; v.w *= sc;
  ((float4*)(xp + (size_t)wave * D))[lane] = v;
  ((float4*)(agg2 + (size_t)wave * D))[lane] = make_float4(0.f, 0.f, 0.f, 0.f);
}

// ---------------- Stage 9: edge filter + pooled message scatter --------------
__global__ void k_edge_out(const long long* __restrict__ ei, const int* __restrict__ node_map,
                           const float* __restrict__ xp, float* __restrict__ agg2,
                           float* __restrict__ out_ns, float* __restrict__ out_nd,
                           float* __restrict__ out_mask, int E) {
  int wave = (int)(((size_t)blockIdx.x * blockDim.x + threadIdx.x) >> 5);
  int lane = threadIdx.x & 31;
  if (wave >= E) return;
  int ns = node_map[(int)ei[wave]];
  int nd = node_map[(int)ei[wave + E]];
  bool m = (ns >= 0) && (nd >= 0);
  if (lane == 0) {
    out_ns[wave]   = m ? (float)ns : 0.f;
    out_nd[wave]   = m ? (float)nd : 0.f;
    out_mask[wave] = m ? 1.f : 0.f;
  }
  if (m) {
    float4 v = ((const float4*)(xp + (size_t)ns * D))[lane];
    float* dst = agg2 + (size_t)nd * D + lane * 4;
    atomicAdd(dst + 0, v.x);
    atomicAdd(dst + 1, v.y);
    atomicAdd(dst + 2, v.z);
    atomicAdd(dst + 3, v.w);
  }
}

// ---------------- Stage 10: x_out = agg2*Wrel^T + x_p*Wroot^T + b (f32 WMMA) -
// One wave computes TWO 16x16 m-tiles against one n-tile so each weight (B)
// load feeds two v_wmma_f32_16x16x4_f32 ops. Per ISA 7.12.2 (32-bit A 16x4):
//   lane L: M(or N) = L%16 ; VGPR pair holds K = k0 + 2*(L/16) + {0,1}
// C/D: VGPR r -> lanes 0-15: M=r, lanes 16-31: M=r+8 ; N = L%16.
__device__ __forceinline__ void store_tile(float* __restrict__ xout, const v8f& c,
                                           int m0, int n, float bias, int k) {
  if (m0 + 16 <= k) {               // fast path (per half-wave): all rows in range
#pragma unroll
    for (int r2 = 0; r2 < 8; ++r2)
      xout[(size_t)(m0 + r2) * D + n] = c[r2] + bias;
  } else {                          // ragged tail tile
#pragma unroll
    for (int r2 = 0; r2 < 8; ++r2)
      if (m0 + r2 < k) xout[(size_t)(m0 + r2) * D + n] = c[r2] + bias;
  }
}

__global__ void k_gemm_wmma(const float* __restrict__ agg2, const float* __restrict__ xp,
                            const float* __restrict__ w_rel, const float* __restrict__ w_root,
                            const float* __restrict__ b_out, float* __restrict__ xout,
                            int k, int numTilesM) {
  int wave = (int)(((size_t)blockIdx.x * blockDim.x + threadIdx.x) >> 5);
  int lane = threadIdx.x & 31;
  const int tilesN = D / 16;                         // 8
  int pairTiles = (numTilesM + 1) / 2;
  if (wave >= pairTiles * tilesN) return;            // wave-uniform exit (EXEC all-ones)
  int tp = wave / tilesN, tn = wave % tilesN;
  int tm0 = tp * 2, tm1 = tp * 2 + 1;
  bool has1 = (tm1 < numTilesM);                     // wave-uniform
  int m0 = tm0 * 16, m1 = tm1 * 16, n0 = tn * 16;
  int half = lane >> 4, lid = lane & 15;

  int mr0 = m0 + lid; if (mr0 >= k) mr0 = k - 1;     // clamp reads for ragged tail
  int mr1 = has1 ? (m1 + lid) : mr0; if (mr1 >= k) mr1 = k - 1;

  const float* arow0  = agg2 + (size_t)mr0 * D;
  const float* xrow0  = xp   + (size_t)mr0 * D;
  const float* arow1  = agg2 + (size_t)mr1 * D;
  const float* xrow1  = xp   + (size_t)mr1 * D;
  const float* wreln  = w_rel  + (size_t)(n0 + lid) * D;   // B(K,N=n) = W[n][K]
  const float* wrootn = w_root + (size_t)(n0 + lid) * D;

  v8f c0 = {0.f, 0.f, 0.f, 0.f, 0.f, 0.f, 0.f, 0.f};
  v8f c1 = {0.f, 0.f, 0.f, 0.f, 0.f, 0.f, 0.f, 0.f};
#pragma unroll 4
  for (int k0 = 0; k0 < D; k0 += 4) {
    int kk = k0 + 2 * half;
    v2f brel  = *(const v2f*)(wreln + kk);     // each B pair feeds two WMMAs
    v2f broot = *(const v2f*)(wrootn + kk);
    v2f a;
    a = *(const v2f*)(arow0 + kk);
    c0 = __builtin_amdgcn_wmma_f32_16x16x4_f32(false, a, false, brel,  (short)0, c0, false, false);
    a = *(const v2f*)(xrow0 + kk);
    c0 = __builtin_amdgcn_wmma_f32_16x16x4_f32(false, a, false, broot, (short)0, c0, false, false);
    a = *(const v2f*)(arow1 + kk);
    c1 = __builtin_amdgcn_wmma_f32_16x16x4_f32(false, a, false, brel,  (short)0, c1, false, false);
    a = *(const v2f*)(xrow1 + kk);
    c1 = __builtin_amdgcn_wmma_f32_16x16x4_f32(false, a, false, broot, (short)0, c1, false, false);
  }

  float bias = b_out[n0 + lid];
  int n = n0 + lid;
  store_tile(xout, c0, m0 + 8 * half, n, bias, k);
  if (has1) store_tile(xout, c1, m1 + 8 * half, n, bias, k);
}

// ============================ host-side orchestration ========================
extern "C" void kernel_launch(void* const* d_in, const int* in_sizes, int n_in,
                              void* d_out, int out_size, void* d_ws, size_t ws_size,
                              hipStream_t stream) {
  const float*      x        = (const float*)d_in[0];
  const long long*  ei       = (const long long*)d_in[1];
  const float*      w_rel_in = (const float*)d_in[2];
  const float*      w_root_in= (const float*)d_in[3];
  const float*      b_in     = (const float*)d_in[4];
  const float*      w_sel    = (const float*)d_in[5];
  const float*      w_rel_out= (const float*)d_in[6];
  const float*      w_root_out=(const float*)d_in[7];
  const float*      b_out    = (const float*)d_in[8];

  const int N = in_sizes[0] / D;
  const int E = in_sizes[1] / 2;
  const int k = (N + 1) / 2;                       // ceil(0.5*N)
  int SORTN = 1; while (SORTN < N) SORTN <<= 1;

  // ---- workspace layout (256B aligned slabs) ----
  char* base = (char*)d_ws;
  size_t off = 0;
  auto alloc = [&](size_t bytes) { char* q = base + off; off = (off + bytes + 255) & ~(size_t)255; return q; };
  unsigned long long* keys = (unsigned long long*)alloc((size_t)SORTN * 8);
  float* p       = (float*)alloc((size_t)N * 4);
  float* r       = (float*)alloc((size_t)N * 4);
  float* attnsum = (float*)alloc((size_t)N * 4);
  float* deg     = (float*)alloc((size_t)N * 4);
  float* attn    = (float*)alloc((size_t)N * 4);
  float* s       = (float*)alloc((size_t)N * 4);
  int*   node_map= (int*)alloc((size_t)N * 4);
  int*   perm_i  = (int*)alloc((size_t)k * 4);
  float* part    = (float*)alloc(4096 * 4);
  float* gmax    = (float*)alloc(256);
  float* gsum    = (float*)alloc(256);
  float* xp      = (float*)alloc((size_t)k * D * 4);
  float* agg2    = (float*)alloc((size_t)k * D * 4);

  // ---- output layout (flat floats, reference tuple order) ----
  float* out_x     = (float*)d_out;                 // [k,128]
  float* out_ns    = out_x    + (size_t)k * D;      // [E]
  float* out_nd    = out_ns   + (size_t)E;          // [E]
  float* out_mask  = out_nd   + (size_t)E;          // [E]
  float* out_batch = out_mask + (size_t)E;          // [k]
  float* out_perm  = out_batch+ (size_t)k;          // [k]
  float* out_score = out_perm + (size_t)k;          // [k]

  const int wavesPerBlk = TB / 32;

  // 1) per-node dots (+ zero attnsum/deg)
  k_node_dots<<<(N + wavesPerBlk - 1) / wavesPerBlk, TB, 0, stream>>>(
      x, w_rel_in, w_root_in, p, r, attnsum, deg, N);

  // 2) scalar edge scatter
  k_edge_scatter_in<<<(E + TB - 1) / TB, TB, 0, stream>>>(ei, p, attnsum, deg, E);

  // 3) attn + softmax max
  int nb = (N + TB - 1) / TB;
  k_attn_max<<<nb, TB, 0, stream>>>(attnsum, r, b_in, attn, part, N);
  k_reduce_final<<<1, TB, 0, stream>>>(part, gmax, nb, 1);

  // 4) exp + softmax sum
  k_exp_sum<<<nb, TB, 0, stream>>>(attn, gmax, part, N);
  k_reduce_final<<<1, TB, 0, stream>>>(part, gsum, nb, 0);

  // 5) scores + sort keys
  k_score_keys<<<(SORTN + TB - 1) / TB, TB, 0, stream>>>(
      attn, deg, gsum, w_sel, s, keys, node_map, N, SORTN);

  // 6) bitonic sort (ascending on flipped key == descending score, stable ties)
  for (int sz = 2; sz <= SORTN; sz <<= 1)
    for (int j = sz >> 1; j > 0; j >>= 1)
      k_bitonic<<<SORTN / TB, TB, 0, stream>>>(keys, j, sz);

  // 7) top-k selection outputs
  k_select<<<(k + TB - 1) / TB, TB, 0, stream>>>(
      keys, s, node_map, perm_i, out_perm, out_score, out_batch, k);

  // 8) pooled features + zero agg2
  k_xp<<<(k + wavesPerBlk - 1) / wavesPerBlk, TB, 0, stream>>>(x, perm_i, s, xp, agg2, k);

  // 9) edge filter + message scatter
  k_edge_out<<<(E + wavesPerBlk - 1) / wavesPerBlk, TB, 0, stream>>>(
      ei, node_map, xp, agg2, out_ns, out_nd, out_mask, E);

  // 10) f32 WMMA GEMM: x_out = agg2*Wrel^T + x_p*Wroot^T + b_out
  int numTilesM = (k + 15) / 16;
  int pairTiles = (numTilesM + 1) / 2;
  int waves = pairTiles * (D / 16);
  k_gemm_wmma<<<(waves + wavesPerBlk - 1) / wavesPerBlk, TB, 0, stream>>>(
      agg2, xp, w_rel_out, w_root_out, b_out, out_x, k, numTilesM);
}